// LRP_PURE_layer_66108136620348
// MI455X (gfx1250) — compile-verified
//
#include <hip/hip_runtime.h>

// Problem constants (from the reference)
#define NN 100000          // nodes
#define EE 400000          // edges
#define PP 250000          // permutations
#define LL 16
#define DD 16
#define TILES (PP / 16)    // 15625 WMMA row-tiles

typedef __attribute__((ext_vector_type(2))) float v2f;
typedef __attribute__((ext_vector_type(8))) float v8f;

// ---------------------------------------------------------------------------
// Main fused kernel: gather -> f32 WMMA GEMM ([16,256]x[256,16]) -> relu ->
// atomic scatter into pooled output.  One wave32 per tile of 16 permutations.
// ---------------------------------------------------------------------------
__global__ __launch_bounds__(256) void lrp_main_kernel(
    const float* __restrict__ nfeat,      // [N,16]
    const int*   __restrict__ efeat_idx,  // [E]
    const int*   __restrict__ n_col,      // [M]
    const float* __restrict__ n_val,      // [M]
    const int*   __restrict__ e_col,      // [M]
    const float* __restrict__ e_val,      // [M]
    const int*   __restrict__ p_row,      // [P]
    const float* __restrict__ p_val,      // [P]
    const float* __restrict__ weights,    // [D,D,L] = d*256 + c*16 + l
    const float* __restrict__ bias,       // [16]
    const float* __restrict__ bond_emb,   // [4,16]
    float*       __restrict__ out)        // [N,16] accumulated atomically
{
    // B matrix in LDS: Bsm[k*16 + c], k = l*16 + d  ->  weights[d, c, l]
    __shared__ float Bsm[256 * 16];
    __shared__ float bondsm[64];

    const int tid = threadIdx.x;
    for (int i = tid; i < 256 * 16; i += 256) {
        const int c = i & 15;
        const int k = i >> 4;
        const int d = k & 15;
        const int l = k >> 4;
        Bsm[i] = weights[d * 256 + c * 16 + l];
    }
    if (tid < 64) bondsm[tid] = bond_emb[tid];
    __syncthreads();

    const int wave = tid >> 5;
    const int lane = tid & 31;
    const int tile = blockIdx.x * 8 + wave;
    if (tile >= TILES) return;           // whole wave exits -> EXEC all-1 for WMMA

    const int row = lane & 15;           // A-matrix row (local permutation)
    const int o   = lane >> 4;           // lane half: selects K +0/+1 vs +2/+3
    const int c   = row;                 // B/C column index for this lane

    v8f acc = {0.f, 0.f, 0.f, 0.f, 0.f, 0.f, 0.f, 0.f};

    const int mbase = tile * 256 + row * 16;   // m = (tile*16 + row)*16 + l

    #pragma unroll 4
    for (int l = 0; l < 16; ++l) {
        const int   m    = mbase + l;
        const int   ncol = n_col[m];
        const float nv   = n_val[m];
        const int   ecol = e_col[m];
        const float ev   = e_val[m];
        const int   bidx = efeat_idx[ecol];

        const float* __restrict__ nf = nfeat + (size_t)ncol * 16;
        const float* __restrict__ be = bondsm + bidx * 16;
        const int k0 = l * 16;

        #pragma unroll
        for (int j = 0; j < 4; ++j) {
            // A fragment: lane needs consecutive pair at d = 4j + 2*o
            const int d = 4 * j + 2 * o;
            const float2 nf2 = *reinterpret_cast<const float2*>(nf + d);
            v2f a;
            a.x = nv * nf2.x + ev * be[d];
            a.y = nv * nf2.y + ev * be[d + 1];
            // B fragment: K rows k0+4j+2o and +1, column c
            const int kk = k0 + 4 * j + 2 * o;
            v2f b;
            b.x = Bsm[kk * 16 + c];
            b.y = Bsm[(kk + 1) * 16 + c];
            acc = __builtin_amdgcn_wmma_f32_16x16x4_f32(
                /*neg_a=*/false, a, /*neg_b=*/false, b,
                /*c_mod=*/(short)0, acc, /*reuse_a=*/false, /*reuse_b=*/false);
        }
    }

    // relu(acc + bias), then pooled[p_row[p], c] += p_val[p] * value
    const float bv = bias[c];
    #pragma unroll
    for (int i = 0; i < 8; ++i) {
        const int pl = i + 8 * o;             // C/D layout: M = i + 8*(lane>=16)
        const int p  = tile * 16 + pl;
        float v = acc[i] + bv;
        v = v > 0.f ? v : 0.f;
        const int   pr = p_row[p];
        const float pv = p_val[p];
        atomicAdd(out + (size_t)pr * 16 + c, pv * v);
    }
}

// ---------------------------------------------------------------------------
// Epilogue: per-node degree-gate MLP (1 -> 32 -> 16) and multiply into out.
// ---------------------------------------------------------------------------
__global__ __launch_bounds__(256) void lrp_fd_kernel(
    const float* __restrict__ degs,  // [N]
    const float* __restrict__ W0,    // [32]
    const float* __restrict__ b0,    // [32]
    const float* __restrict__ W1,    // [32,16]
    const float* __restrict__ b1,    // [16]
    float*       __restrict__ out)   // [N,16]
{
    __shared__ float sW0[32], sb0[32], sW1[512], sb1[16];
    const int tid = threadIdx.x;
    if (tid < 32) { sW0[tid] = W0[tid]; sb0[tid] = b0[tid]; }
    if (tid < 16) sb1[tid] = b1[tid];
    for (int i = tid; i < 512; i += 256) sW1[i] = W1[i];
    __syncthreads();

    const long long t = (long long)blockIdx.x * 256 + tid;
    if (t >= (long long)NN * 16) return;
    const int n = (int)(t >> 4);
    const int c = (int)(t & 15);

    const float deg = degs[n];
    float fd = sb1[c];
    #pragma unroll 8
    for (int h = 0; h < 32; ++h) {
        float hv = deg * sW0[h] + sb0[h];
        hv = hv > 0.f ? hv : 0.f;
        fd += hv * sW1[h * 16 + c];
    }
    out[t] *= fd;
}

// ---------------------------------------------------------------------------
extern "C" void kernel_launch(void* const* d_in, const int* in_sizes, int n_in,
                              void* d_out, int out_size, void* d_ws, size_t ws_size,
                              hipStream_t stream)
{
    (void)in_sizes; (void)n_in; (void)out_size; (void)d_ws; (void)ws_size;

    const float* nfeat     = (const float*)d_in[0];
    const int*   efeat_idx = (const int*)  d_in[1];
    // d_in[2] = n_row (arange, unused)
    const int*   n_col     = (const int*)  d_in[3];
    const float* n_val     = (const float*)d_in[4];
    // d_in[5] = e_row (arange, unused)
    const int*   e_col     = (const int*)  d_in[6];
    const float* e_val     = (const float*)d_in[7];
    const int*   p_row     = (const int*)  d_in[8];
    // d_in[9] = p_col (arange, unused)
    const float* p_val     = (const float*)d_in[10];
    const float* degs      = (const float*)d_in[11];
    const float* weights   = (const float*)d_in[12];
    const float* bias      = (const float*)d_in[13];
    const float* W0        = (const float*)d_in[14];
    const float* b0        = (const float*)d_in[15];
    const float* W1        = (const float*)d_in[16];
    const float* b1        = (const float*)d_in[17];
    const float* bond_emb  = (const float*)d_in[18];
    float* out = (float*)d_out;

    // Zero the accumulation target (graph-capture-safe async memset).
    hipMemsetAsync(out, 0, (size_t)NN * 16 * sizeof(float), stream);

    const int blocks = (TILES + 7) / 8;   // 8 waves (tiles) per 256-thread block
    lrp_main_kernel<<<blocks, 256, 0, stream>>>(
        nfeat, efeat_idx, n_col, n_val, e_col, e_val,
        p_row, p_val, weights, bias, bond_emb, out);

    const long long total = (long long)NN * 16;
    lrp_fd_kernel<<<(int)((total + 255) / 256), 256, 0, stream>>>(
        degs, W0, b0, W1, b1, out);
}